// FLandMDCA_19370302505551
// MI455X (gfx1250) — compile-verified
//
#include <hip/hip_runtime.h>
#include <hip/hip_bf16.h>

// FocalLoss + MDCA for probs[16384,1000] f32, targets[16384] int.
// Bandwidth-bound (65.5 MB read, ~2.8us floor @ 23.3 TB/s).
// Column sums done as ones(16x4) @ probs-tile(4x16) via V_WMMA_F32_16X16X4_F32.

#define B_DIM 16384
#define C_DIM 1000
#define ROWS_PER_BLOCK 64
#define NBLOCKS (B_DIM / ROWS_PER_BLOCK)   // 256
#define NTILES 63                          // ceil(1000 / 16)
#define WAVES_PER_BLOCK 8

typedef __attribute__((ext_vector_type(2))) float v2f;
typedef __attribute__((ext_vector_type(8))) float v8f;

__global__ __launch_bounds__(256) void fl_mdca_partials(
    const float* __restrict__ probs,
    const int*   __restrict__ targets,
    float*       __restrict__ colsum_ws,   // [NBLOCKS][C_DIM]
    float*       __restrict__ focal_ws,    // [NBLOCKS]
    int*         __restrict__ counts)      // [C_DIM], pre-zeroed
{
    const int block = blockIdx.x;
    const int tid   = threadIdx.x;
    const int wave  = tid >> 5;
    const int lane  = tid & 31;
    const int row0  = block * ROWS_PER_BLOCK;

    // ---- Phase 1: partial column sums over 64 rows via WMMA -------------
    // A = ones(16x4)  =>  D[m,n] += sum_k B[k,n]; all D rows identical.
    // Lane layout for B (4x16 f32): VGPR0 holds K=2*kh, VGPR1 holds K=2*kh+1,
    // n = lane&15, kh = lane>>4. (Exact K mapping is immaterial: A is ones.)
    const v2f a_ones = {1.0f, 1.0f};

    const int n    = lane & 15;
    const int kh   = lane >> 4;

    for (int tile = wave; tile < NTILES; tile += WAVES_PER_BLOCK) {
        const int col   = tile * 16 + n;
        const bool valid = (col < C_DIM);
        const int colc  = valid ? col : (C_DIM - 1);   // clamped safe address

        v8f acc = {};
        #pragma unroll
        for (int r = 0; r < ROWS_PER_BLOCK; r += 4) {
            const float* base = probs + (size_t)(row0 + r + 2 * kh) * C_DIM + colc;
            v2f b;
            float b0 = base[0];        // K = 2*kh
            float b1 = base[C_DIM];    // K = 2*kh + 1
            b.x = valid ? b0 : 0.0f;   // cndmask, EXEC stays all-ones for WMMA
            b.y = valid ? b1 : 0.0f;
            acc = __builtin_amdgcn_wmma_f32_16x16x4_f32(
                false, a_ones, false, b, (short)0, acc, false, false);
        }
        // acc[0] on lanes 0..15 = column sum for column tile*16+lane (M=0 row)
        if (lane < 16 && valid) {
            colsum_ws[(size_t)block * C_DIM + col] = acc[0];
        }
    }

    // ---- Phase 2: focal partial + target histogram ----------------------
    __shared__ float sred[ROWS_PER_BLOCK];
    if (tid < ROWS_PER_BLOCK) {
        const int row = row0 + tid;
        const int t   = targets[row];
        const float p = probs[(size_t)row * C_DIM + t];
        const float lp = logf(p);
        sred[tid] = -(1.0f - p) * lp;          // gamma = 1
        atomicAdd(&counts[t], 1);              // integer atomic: deterministic
    }
    __syncthreads();
    if (tid < 32) {
        float v = sred[tid] + sred[tid + 32];
        #pragma unroll
        for (int off = 16; off; off >>= 1) v += __shfl_down(v, off, 32);
        if (tid == 0) focal_ws[block] = v;
    }
}

__global__ __launch_bounds__(1024) void fl_mdca_final(
    const float* __restrict__ colsum_ws,
    const float* __restrict__ focal_ws,
    const int*   __restrict__ counts,
    float*       __restrict__ out)
{
    const int tid = threadIdx.x;

    // Calibration term: fold 256 block-partials per column (fixed order ->
    // deterministic). Consecutive lanes read consecutive columns: coalesced.
    float cal = 0.0f;
    if (tid < C_DIM) {
        float s = 0.0f;
        for (int b = 0; b < NBLOCKS; ++b)
            s += colsum_ws[(size_t)b * C_DIM + tid];
        const float avg_conf  = s * (1.0f / (float)B_DIM);
        const float avg_count = (float)counts[tid] * (1.0f / (float)B_DIM);
        cal = fabsf(avg_conf - avg_count);
    }

    float foc = (tid < NBLOCKS) ? focal_ws[tid] : 0.0f;

    __shared__ float scal[32];
    __shared__ float sfoc[32];
    const int wave = tid >> 5;
    const int lane = tid & 31;
    #pragma unroll
    for (int off = 16; off; off >>= 1) {
        cal += __shfl_down(cal, off, 32);
        foc += __shfl_down(foc, off, 32);
    }
    if (lane == 0) { scal[wave] = cal; sfoc[wave] = foc; }
    __syncthreads();
    if (tid < 32) {   // 1024 threads = exactly 32 waves
        cal = scal[tid];
        foc = sfoc[tid];
        #pragma unroll
        for (int off = 16; off; off >>= 1) {
            cal += __shfl_down(cal, off, 32);
            foc += __shfl_down(foc, off, 32);
        }
        if (tid == 0)
            out[0] = foc * (1.0f / (float)B_DIM) + cal * (1.0f / (float)C_DIM);
    }
}

extern "C" void kernel_launch(void* const* d_in, const int* in_sizes, int n_in,
                              void* d_out, int out_size, void* d_ws, size_t ws_size,
                              hipStream_t stream) {
    const float* probs   = (const float*)d_in[0];
    const int*   targets = (const int*)d_in[1];
    float*       out     = (float*)d_out;

    float* colsum_ws = (float*)d_ws;                       // NBLOCKS*C_DIM floats
    float* focal_ws  = colsum_ws + (size_t)NBLOCKS * C_DIM; // NBLOCKS floats
    int*   counts    = (int*)(focal_ws + NBLOCKS);          // C_DIM ints

    hipMemsetAsync(counts, 0, C_DIM * sizeof(int), stream);

    fl_mdca_partials<<<NBLOCKS, 256, 0, stream>>>(probs, targets,
                                                  colsum_ws, focal_ws, counts);
    fl_mdca_final<<<1, 1024, 0, stream>>>(colsum_ws, focal_ws, counts, out);
}